// SumPool_5325759447404
// MI455X (gfx1250) — compile-verified
//
#include <hip/hip_runtime.h>
#include <stdint.h>

// Problem sizes (fixed by the reference).
#define N_ATOMS_C 4194304
#define N_MOLS_C  32768

#define THREADS          256
#define TILE_ATOMS       2048
#define TILE_FLOATS      (TILE_ATOMS * 3)          // 6144 floats
#define TILE_BYTES       (TILE_FLOATS * 4)         // 24576 B (16B-aligned)
#define TILES_PER_BLOCK  4
#define NUM_BLOCKS       (N_ATOMS_C / (TILE_ATOMS * TILES_PER_BLOCK))  // 512

// The async-to-LDS builtin takes pointers to 16-byte int vectors.
typedef int v4i __attribute__((vector_size(16)));
typedef __attribute__((address_space(1))) v4i* gv4_ptr;   // global src
typedef __attribute__((address_space(3))) v4i* lv4_ptr;   // LDS dst

// ---- CDNA5 async global->LDS copy of one wave's 3KB slice of a tile ------
// Per wave: 6 x global_load_async_to_lds_b128 (32 lanes x 16B = 512B each).
__device__ __forceinline__ void async_tile_load(const float* xyz,
                                                unsigned lds_base,
                                                size_t tile_byte, int tid) {
  const unsigned wave = (unsigned)tid >> 5;
  const unsigned lane = (unsigned)tid & 31u;
#pragma unroll
  for (int i = 0; i < 6; ++i) {
    const unsigned o        = wave * 3072u + (unsigned)i * 512u + lane * 16u;
    const unsigned lds_addr = lds_base + o;
    const uint64_t gaddr    = (uint64_t)(uintptr_t)xyz + (uint64_t)tile_byte + o;
#if __has_builtin(__builtin_amdgcn_global_load_async_to_lds_b128)
    __builtin_amdgcn_global_load_async_to_lds_b128(
        (gv4_ptr)(uintptr_t)gaddr, (lv4_ptr)(uintptr_t)lds_addr, 0, 0);
#else
    asm volatile("global_load_async_to_lds_b128 %0, %1, off"
                 :
                 : "v"(lds_addr), "v"(gaddr)
                 : "memory");
#endif
  }
}

__device__ __forceinline__ void wait_async_le6() {
#if __has_builtin(__builtin_amdgcn_s_wait_asynccnt)
  __builtin_amdgcn_s_wait_asynccnt(6);
#else
  asm volatile("s_wait_asynccnt 6" ::: "memory");
#endif
}
__device__ __forceinline__ void wait_async_le0() {
#if __has_builtin(__builtin_amdgcn_s_wait_asynccnt)
  __builtin_amdgcn_s_wait_asynccnt(0);
#else
  asm volatile("s_wait_asynccnt 0" ::: "memory");
#endif
}

// Fused forward+backward: energy[mol] += tanh(x.w); grad = (1-t^2)*w.
__global__ __launch_bounds__(THREADS) void molpool_fused_kernel(
    const float* __restrict__ xyz, const int* __restrict__ seg,
    const float* __restrict__ w, float* __restrict__ energy,
    float* __restrict__ grad) {
  __shared__ __align__(16) float lbuf[2][TILE_FLOATS];

  const int tid  = threadIdx.x;
  const int lane = tid & 31;
  const float wx = w[0], wy = w[1], wz = w[2];

  const size_t firstTile = (size_t)blockIdx.x * TILES_PER_BLOCK;
  const unsigned lds_b0 = (unsigned)(uintptr_t)&lbuf[0][0];
  const unsigned lds_b1 = (unsigned)(uintptr_t)&lbuf[1][0];

  // Prime the double buffer.
  async_tile_load(xyz, lds_b0, (firstTile + 0) * (size_t)TILE_BYTES, tid);
  async_tile_load(xyz, lds_b1, (firstTile + 1) * (size_t)TILE_BYTES, tid);

  for (int t = 0; t < TILES_PER_BLOCK; ++t) {
    // ASYNCcnt decrements in order: <=6 means everything but the newest
    // 6-instruction batch (tile t+1) has landed; last tile has no successor.
    if (t < TILES_PER_BLOCK - 1) wait_async_le6(); else wait_async_le0();
    __syncthreads();  // all waves' slices of tile t are now in LDS

    const float* buf = lbuf[t & 1];
    const int tile_atom0 = (int)(firstTile + (size_t)t) * TILE_ATOMS;

#pragma unroll
    for (int i = 0; i < TILE_ATOMS / THREADS; ++i) {
      const int a    = i * THREADS + tid;     // lanes hold consecutive atoms
      const int atom = tile_atom0 + a;

      // Conflict-free LDS reads: dword bank = 3*tid mod 64 (bijective).
      const float x = buf[3 * a + 0];
      const float y = buf[3 * a + 1];
      const float z = buf[3 * a + 2];

      const float d  = fmaf(x, wx, fmaf(y, wy, z * wz));
      const float th = tanhf(d);
      const float g  = 1.0f - th * th;

      // Write-once 48MB stream: non-temporal, keep L2 for energy atomics.
      float* gp = grad + (size_t)atom * 3;
      __builtin_nontemporal_store(g * wx, gp + 0);
      __builtin_nontemporal_store(g * wy, gp + 1);
      __builtin_nontemporal_store(g * wz, gp + 2);

      // Sorted segment ids: capture the wave's run structure with ONE shfl +
      // one ballot (head flags), then a 5-step val-only segmented scan.
      const int s = __builtin_nontemporal_load(seg + atom);
      const int sprev = __shfl_up(s, 1, 32);
      const bool head = (lane == 0) || (sprev != s);
      const unsigned hb = (unsigned)__ballot(head);          // wave32 mask

      // First lane of my run = highest head bit at position <= lane.
      const unsigned below = hb & (0xFFFFFFFFu >> (31 - lane));
      const int start = 31 - __clz(below);                   // below != 0

      float val = th;
#pragma unroll
      for (int dlt = 1; dlt < 32; dlt <<= 1) {
        const float uv = __shfl_up(val, dlt, 32);
        if (lane - dlt >= start) val += uv;                  // stay inside run
      }
      // Tail of run: last lane, or next lane starts a new run.
      const bool tail = (lane == 31) || ((hb >> (lane + 1)) & 1u);
      if (tail)
        unsafeAtomicAdd(energy + s, val);                    // global_atomic_add_f32
    }

    __syncthreads();  // everyone done reading buf[t&1] before refill
    if (t + 2 < TILES_PER_BLOCK)
      async_tile_load(xyz, (t & 1) ? lds_b1 : lds_b0,
                      (firstTile + (size_t)t + 2) * (size_t)TILE_BYTES, tid);
  }
}

extern "C" void kernel_launch(void* const* d_in, const int* in_sizes, int n_in,
                              void* d_out, int out_size, void* d_ws, size_t ws_size,
                              hipStream_t stream) {
  (void)in_sizes; (void)n_in; (void)out_size; (void)d_ws; (void)ws_size;
  const float* xyz = (const float*)d_in[0];
  const int*   seg = (const int*)d_in[1];
  const float* w   = (const float*)d_in[2];
  float* energy = (float*)d_out;           // [N_MOLS]
  float* grad   = energy + N_MOLS_C;       // [N_ATOMS, 3]

  (void)hipMemsetAsync(energy, 0, N_MOLS_C * sizeof(float), stream);
  hipLaunchKernelGGL(molpool_fused_kernel, dim3(NUM_BLOCKS), dim3(THREADS), 0,
                     stream, xyz, seg, w, energy, grad);
}